// SubShiftedLoss_5231270167089
// MI455X (gfx1250) — compile-verified
//
#include <hip/hip_runtime.h>
#include <hip/hip_bf16.h>

typedef float v2f __attribute__((ext_vector_type(2)));
typedef float v8f __attribute__((ext_vector_type(8)));

#define CCH   64
#define HH    192
#define WW    192
#define SS    16
#define NBD   12
#define NSH   7           // shifts per dim (-3..3)
#define NCHK  6           // ceil(22/4) K-chunks of 4
#define CPP   32          // channels staged per pass (32KB LDS)
#define PLANE (HH * WW)

// One workgroup per (batch, block). 4 waves, each owns 1/4 of channels,
// two f32 WMMA accumulator chains: Cross (fm2 x Toeplitz(blk)) and
// Sb (fm2^2 x Toeplitz(ones)).
__global__ __launch_bounds__(128) void ssl_block_min_kernel(
    const float* __restrict__ fm1,
    const float* __restrict__ fm2,
    float* __restrict__ minpb)
{
    __shared__ float lds_blk[CPP * SS * SS];    // 32 KB staged fm1 block (half)
    __shared__ float lds_red[2 * 4 * 32 * 8];   // 8 KB accumulator dump
    __shared__ float lds_sa[128];
    __shared__ float lds_min[64];

    const int wg   = blockIdx.x;
    const int b    = wg / (NBD * NBD);
    const int blkI = wg % (NBD * NBD);
    const int by   = blkI / NBD;
    const int bx   = blkI % NBD;
    const int tid  = threadIdx.x;
    const int wave = tid >> 5;
    const int lane = tid & 31;

    const float* f1b = fm1 + (size_t)b * CCH * PLANE;
    const float* f2b = fm2 + (size_t)b * CCH * PLANE;

    // ---- per-lane, loop-invariant WMMA operand addressing ----
    // A (16x4 f32): lane&15 = M row, lane>>4 selects K pair {k0,k0+1} vs {k0+2,k0+3}
    // B (4x16 f32): lane&15 = N col, same K-pair split (mirror of A layout)
    const int nIdx  = lane & 15;
    const int khalf = lane >> 4;
    const int mA    = nIdx < (NSH - 1) ? nIdx : (NSH - 1);   // dh row (rows 7..15 unused)
    int rowb = by * SS + mA - 3;                             // clip like reference y0
    rowb = rowb < 0 ? 0 : (rowb > HH - SS ? HH - SS : rowb);
    const int xb = bx * SS - 3;                              // patch column origin

    int aoff0[NCHK], aoff1[NCHK], boff0[NCHK], boff1[NCHK];
    unsigned msk0[NCHK], msk1[NCHK];
    float one0[NCHK], one1[NCHK];
#pragma unroll
    for (int kc = 0; kc < NCHK; ++kc) {
        const int k0 = kc * 4 + 2 * khalf;
        const int k1 = k0 + 1;
        // A columns: clamp each element separately (clamped ones only feed
        // edge-invalid shift columns, which get masked to +inf below)
        int c0 = xb + k0; c0 = c0 < 0 ? 0 : (c0 > WW - 1 ? WW - 1 : c0);
        int c1 = xb + k1; c1 = c1 < 0 ? 0 : (c1 > WW - 1 ? WW - 1 : c1);
        aoff0[kc] = (rowb * WW + c0) * 4;
        aoff1[kc] = (rowb * WW + c1) * 4;
        // B Toeplitz: B[k, n] = blkrow[k - n] if 0 <= k-n < 16 and n < 7
        const int i0 = k0 - nIdx;
        const int i1 = k1 - nIdx;
        const bool v0 = (nIdx < NSH) && (i0 >= 0) && (i0 < SS);
        const bool v1 = (nIdx < NSH) && (i1 >= 0) && (i1 < SS);
        int j0 = i0 < 0 ? 0 : (i0 > SS - 1 ? SS - 1 : i0);
        int j1 = i1 < 0 ? 0 : (i1 > SS - 1 ? SS - 1 : i1);
        boff0[kc] = j0;
        boff1[kc] = j1;
        msk0[kc] = v0 ? 0xffffffffu : 0u;
        msk1[kc] = v1 ? 0xffffffffu : 0u;
        one0[kc] = v0 ? 1.0f : 0.0f;
        one1[kc] = v1 ? 1.0f : 0.0f;
    }

    float sa_part = 0.0f;
    v8f acc_cr = {0.f, 0.f, 0.f, 0.f, 0.f, 0.f, 0.f, 0.f};
    v8f acc_sb = {0.f, 0.f, 0.f, 0.f, 0.f, 0.f, 0.f, 0.f};

    for (int p = 0; p < CCH / CPP; ++p) {
        // ---- stage fm1 block channels [p*CPP, p*CPP+CPP) into LDS, accumulate Sa
        for (int r = tid; r < CPP * SS; r += 128) {
            const int cc = r >> 4;
            const int i  = r & 15;
            const float4* src = (const float4*)(f1b +
                ((size_t)(p * CPP + cc) * HH + (by * SS + i)) * WW + bx * SS);
            float4* dst = (float4*)&lds_blk[r * SS];
#pragma unroll
            for (int q = 0; q < 4; ++q) {
                float4 v = src[q];
                dst[q] = v;
                sa_part += v.x * v.x + v.y * v.y + v.z * v.z + v.w * v.w;
            }
        }
        __syncthreads();

        // ---- GEMM accumulation over this pass's channels
        for (int c8 = 0; c8 < CPP / 4; ++c8) {
            const int cc = c8 * 4 + wave;
            const char* plane = (const char*)(f2b + (size_t)(p * CPP + cc) * PLANE);
            const float* brow0 = &lds_blk[cc * SS * SS];
            __builtin_prefetch(plane + aoff0[0], 0, 1);   // global_prefetch_b8
            for (int i = 0; i < SS; ++i) {
                const char* arow = plane + i * (WW * 4);
                const float* brow = brow0 + i * SS;
#pragma unroll
                for (int kc = 0; kc < NCHK; ++kc) {
                    v2f a, a2, bt, bo;
                    a.x = *(const float*)(arow + aoff0[kc]);
                    a.y = *(const float*)(arow + aoff1[kc]);
                    a2.x = a.x * a.x;
                    a2.y = a.y * a.y;
                    bt.x = __uint_as_float(__float_as_uint(brow[boff0[kc]]) & msk0[kc]);
                    bt.y = __uint_as_float(__float_as_uint(brow[boff1[kc]]) & msk1[kc]);
                    bo.x = one0[kc];
                    bo.y = one1[kc];
                    acc_cr = __builtin_amdgcn_wmma_f32_16x16x4_f32(
                        false, a,  false, bt, (short)0, acc_cr, false, false);
                    acc_sb = __builtin_amdgcn_wmma_f32_16x16x4_f32(
                        false, a2, false, bo, (short)0, acc_sb, false, false);
                }
            }
        }
        __syncthreads();
    }

    // ---- dump accumulators + reduce Sa
    lds_sa[tid] = sa_part;
    {
        float* rc = &lds_red[(wave * 32 + lane) * 8];
        float* rs = &lds_red[1024 + (wave * 32 + lane) * 8];
#pragma unroll
        for (int r = 0; r < 8; ++r) { rc[r] = acc_cr[r]; rs[r] = acc_sb[r]; }
    }
    __syncthreads();
    for (int s2 = 64; s2 > 0; s2 >>= 1) {
        if (tid < s2) lds_sa[tid] += lds_sa[tid + s2];
        __syncthreads();
    }
    const float sa = lds_sa[0];

    // ---- 49 shift distances, mask edge-invalid shifts, min-reduce
    float myv = __builtin_inff();
    if (tid < NSH * NSH) {
        const int m = tid / NSH;
        const int n = tid % NSH;
        float cr = 0.f, sb = 0.f;
#pragma unroll
        for (int w2 = 0; w2 < 4; ++w2) {
            cr += lds_red[(w2 * 32 + n) * 8 + m];              // C[m, n] lives at vgpr m, lane n
            sb += lds_red[1024 + (w2 * 32 + n) * 8 + m];
        }
        const int dy = by * SS + m - 3;
        const int dx = bx * SS + n - 3;
        const bool valid = (dy >= 0) && (dy <= HH - SS) && (dx >= 0) && (dx <= WW - SS);
        const float dist = (sa - 2.0f * cr + sb) * (1.0f / (float)(CCH * SS * SS));
        myv = valid ? dist : __builtin_inff();
    }
    if (tid < 64) lds_min[tid] = myv;
    __syncthreads();
    for (int s2 = 32; s2 > 0; s2 >>= 1) {
        if (tid < s2) lds_min[tid] = fminf(lds_min[tid], lds_min[tid + s2]);
        __syncthreads();
    }
    if (tid == 0) minpb[wg] = lds_min[0];
}

// Per batch: sum of the 16 smallest of 144 block minima (deterministic rank select).
__global__ __launch_bounds__(256) void ssl_topk_sum_kernel(
    const float* __restrict__ minpb, float* __restrict__ out)
{
    __shared__ float vals[NBD * NBD];
    __shared__ float part[256];
    const int b = blockIdx.x;
    const int tid = threadIdx.x;
    if (tid < NBD * NBD) vals[tid] = minpb[b * NBD * NBD + tid];
    __syncthreads();
    float sel = 0.0f;
    if (tid < NBD * NBD) {
        const float v = vals[tid];
        int rank = 0;
        for (int j = 0; j < NBD * NBD; ++j) {
            const float u = vals[j];
            if (u < v || (u == v && j < tid)) ++rank;
        }
        if (rank < 16) sel = v;
    }
    part[tid] = sel;
    __syncthreads();
    for (int off = 128; off > 0; off >>= 1) {
        if (tid < off) part[tid] += part[tid + off];
        __syncthreads();
    }
    if (tid == 0) out[b] = part[0];
}

extern "C" void kernel_launch(void* const* d_in, const int* in_sizes, int n_in,
                              void* d_out, int out_size, void* d_ws, size_t ws_size,
                              hipStream_t stream) {
    (void)in_sizes; (void)n_in; (void)out_size; (void)ws_size;
    const float* fm1 = (const float*)d_in[0];
    const float* fm2 = (const float*)d_in[1];
    float* minpb = (float*)d_ws;          // 8 * 144 floats of scratch
    float* out   = (float*)d_out;         // 8 floats

    ssl_block_min_kernel<<<dim3(8 * NBD * NBD), dim3(128), 0, stream>>>(fm1, fm2, minpb);
    ssl_topk_sum_kernel<<<dim3(8), dim3(256), 0, stream>>>(minpb, out);
}